// MixtureModelEncoder_74053826118258
// MI455X (gfx1250) — compile-verified
//
#include <hip/hip_runtime.h>
#include <hip/hip_bf16.h>
#include <math.h>

// ---------------------------------------------------------------------------
// MixtureModelEncoder (ECC-GNN) forward for MI455X / gfx1250, wave32 WMMA.
// All dense math via V_WMMA_F32_16X16X4_F32 (exact f32).
// Shapes are compile-time (template): branch-free WMMA k-loops, fixed
// accumulator registers, unguarded stores when M % 16 == 0.
// ---------------------------------------------------------------------------

typedef float v2f __attribute__((ext_vector_type(2)));
typedef float v8f __attribute__((ext_vector_type(8)));

__device__ __forceinline__ v8f wmma4(v8f acc, v2f a, v2f b) {
  return __builtin_amdgcn_wmma_f32_16x16x4_f32(
      /*neg_a=*/false, a, /*neg_b=*/false, b,
      /*c_mod=*/(short)0, acc, /*reuse_a=*/false, /*reuse_b=*/false);
}

#define NNODES 20000
#define NEDGES 40000
#define NGRAPH 1000

// -------------------------- zero fill --------------------------------------
__global__ void k_zero(float* __restrict__ p, int n) {
  int i = blockIdx.x * blockDim.x + threadIdx.x;
  if (i < n) p[i] = 0.0f;
}

// ------------------ generic f32 WMMA GEMM: D = A@B + bias ------------------
// A: MxK row-major, B: KxN row-major, bias: N, D: MxN row-major.
// NSUB = 16-col subtiles per wave, K compile time, MAL = (M % 16 == 0).
// Grid: x = row-tile groups (8 waves/block, 16 rows/wave), y = slab groups.
// B panel (K x 16*NSUB) staged in LDS once per block.
template <int NSUB, int K, bool MAL>
__global__ __launch_bounds__(256) void k_gemm(
    const float* __restrict__ A, const float* __restrict__ B,
    const float* __restrict__ bias, float* __restrict__ D, int M, int N) {
  constexpr int NW = 16 * NSUB;
  __shared__ float Bs[K * NW];
  int tid = threadIdx.x;
  int n0 = blockIdx.y * NW;

  // stage B panel (N is a multiple of NW for every call site)
  for (int idx = tid; idx < K * NW; idx += 256) {
    int k = idx / NW, c = idx - k * NW;
    Bs[idx] = B[(long)k * N + n0 + c];
  }
  __syncthreads();

  int lane = tid & 31, wave = tid >> 5;
  int tiles_m = (M + 15) >> 4;
  int tm = blockIdx.x * 8 + wave;
  if (tm >= tiles_m) return;               // wave-uniform, after the only sync
  int m0 = tm << 4;
  int lo = lane & 15, hi = lane >> 4;

  int arow = m0 + lo;
  if (!MAL) { if (arow >= M) arow = M - 1; }
  const float* __restrict__ arp = A + (long)arow * K;
  __builtin_prefetch(arp, 0, 1);

  v8f z = {0.f,0.f,0.f,0.f,0.f,0.f,0.f,0.f};
  v8f acc[NSUB];
  #pragma unroll
  for (int j = 0; j < NSUB; ++j) acc[j] = z;

  for (int k0 = 0; k0 < K; k0 += 4) {
    int kk = k0 + (hi << 1);
    v2f a;
    a.x = arp[kk];
    a.y = arp[kk + 1];
    const float* __restrict__ br0 = &Bs[kk * NW + lo];
    #pragma unroll
    for (int j = 0; j < NSUB; ++j) {
      v2f b;
      b.x = br0[j * 16];
      b.y = br0[NW + j * 16];
      acc[j] = wmma4(acc[j], a, b);
    }
  }
  #pragma unroll
  for (int j = 0; j < NSUB; ++j) {
    int col = n0 + j * 16 + lo;
    float bb = bias[col];
    #pragma unroll
    for (int r = 0; r < 8; ++r) {
      int m = m0 + r + (hi << 3);
      if (MAL || m < M) D[(long)m * N + col] = acc[j][r] + bb;
    }
  }
}

// ----------- ECC message GEMM: msg = A_aug @ W_aug (fused, E x cout) -------
// A_aug[e, b*cin+i] = (b<4 ? ea[e,b] : 1) * x[src[e], i]    (K = 5*cin)
// W_aug[b*cin+i, o] = (b<4 ? W_edge[b, i*cout+o] : b_edge[i*cout+o])
// Gathered x rows (128/block) staged in LDS once. Since cin % 4 == 0 and
// k0 is 4-aligned, the bond slot bs = k0>>LC is uniform per k-step.
template <int LC, int NSUB>
__global__ __launch_bounds__(256) void k_ecc_msg(
    const float* __restrict__ x, const int* __restrict__ src,
    const float* __restrict__ ea, const float* __restrict__ W_edge,
    const float* __restrict__ b_edge, float* __restrict__ msg,
    int cout, int E) {
  constexpr int CIN = 1 << LC;
  constexpr int K = 5 * CIN;
  __shared__ int   Ss[128];
  __shared__ float Xs[128 * CIN];
  int tid = threadIdx.x;
  int e_base = blockIdx.x << 7;            // 128 edges per block

  if (tid < 128) {
    int e = e_base + tid; if (e >= E) e = E - 1;
    Ss[tid] = src[e];
  }
  __syncthreads();
  for (int idx = tid; idx < 128 * CIN; idx += 256) {
    int r = idx >> LC, i = idx & (CIN - 1);
    Xs[idx] = x[(long)Ss[r] * CIN + i];
  }
  __syncthreads();

  int lane = tid & 31, wave = tid >> 5;
  int tm = blockIdx.x * 8 + wave;
  if (tm >= (E >> 4)) return;              // wave-uniform, after syncs
  int e0 = tm << 4;
  int lo = lane & 15, hi = lane >> 4;

  int e = e0 + lo;
  float w[5];
  w[0] = ea[e * 4 + 0]; w[1] = ea[e * 4 + 1];
  w[2] = ea[e * 4 + 2]; w[3] = ea[e * 4 + 3];
  w[4] = 1.0f;
  const float* __restrict__ xrow = &Xs[((wave << 4) + lo) << LC];

  int n0 = blockIdx.y * (16 * NSUB);
  long bstride = (long)CIN * cout;

  v8f z = {0.f,0.f,0.f,0.f,0.f,0.f,0.f,0.f};
  v8f acc[NSUB];
  #pragma unroll
  for (int j = 0; j < NSUB; ++j) acc[j] = z;

  #pragma unroll 5
  for (int k0 = 0; k0 < K; k0 += 4) {
    int bs = k0 >> LC;                     // uniform over the 4 k's this step
    int i0 = (k0 & (CIN - 1)) + (hi << 1);
    float sc = w[bs];
    v2f a;
    a.x = sc * xrow[i0];
    a.y = sc * xrow[i0 + 1];
    const float* __restrict__ wr =
        (bs < 4) ? (W_edge + (long)bs * bstride) : b_edge;
    const float* __restrict__ wp = wr + (long)i0 * cout + n0 + lo;
    #pragma unroll
    for (int j = 0; j < NSUB; ++j) {
      v2f b;
      b.x = wp[j * 16];
      b.y = wp[cout + j * 16];
      acc[j] = wmma4(acc[j], a, b);
    }
  }
  #pragma unroll
  for (int j = 0; j < NSUB; ++j) {
    int col = n0 + j * 16 + lo;
    #pragma unroll
    for (int r = 0; r < 8; ++r)
      msg[(long)(e0 + r + (hi << 3)) * cout + col] = acc[j][r];
  }
}

// ------------- scatter msg -> agg[dst], deg[dst]  (C = 1<<lC) --------------
__global__ void k_scatter_edges(const float* __restrict__ msg,
                                const int* __restrict__ dst,
                                float* __restrict__ agg, float* __restrict__ deg,
                                int E, int lC) {
  int idx = blockIdx.x * blockDim.x + threadIdx.x;
  if (idx >= (E << lC)) return;
  int e = idx >> lC, c = idx & ((1 << lC) - 1);
  int d = dst[e];
  unsafeAtomicAdd(&agg[(d << lC) + c], msg[idx]);
  if (c == 0) unsafeAtomicAdd(&deg[d], 1.0f);
}

// ------------------ h += agg / max(deg,1)  (C = 1<<lC) ---------------------
__global__ void k_add_agg(float* __restrict__ h, const float* __restrict__ agg,
                          const float* __restrict__ deg, int M, int lC) {
  int idx = blockIdx.x * blockDim.x + threadIdx.x;
  if (idx >= (M << lC)) return;
  int m = idx >> lC;
  h[idx] += agg[idx] / fmaxf(deg[m], 1.0f);
}

// ------------------ batchnorm stats: one block per channel -----------------
__global__ __launch_bounds__(256) void k_bn_stats(
    const float* __restrict__ h, int M, int C,
    float* __restrict__ sums, float* __restrict__ sumsqs) {
  int c = blockIdx.x;
  float s = 0.f, s2 = 0.f;
  for (int m = threadIdx.x; m < M; m += blockDim.x) {
    float v = h[(long)m * C + c];
    s += v; s2 += v * v;
  }
  __shared__ float sh[256], sh2[256];
  sh[threadIdx.x] = s; sh2[threadIdx.x] = s2;
  __syncthreads();
  for (int off = 128; off > 0; off >>= 1) {
    if ((int)threadIdx.x < off) {
      sh[threadIdx.x]  += sh[threadIdx.x + off];
      sh2[threadIdx.x] += sh2[threadIdx.x + off];
    }
    __syncthreads();
  }
  if (threadIdx.x == 0) { sums[c] = sh[0]; sumsqs[c] = sh2[0]; }
}

// ------------------ in-place BN + PReLU  (C = 1<<lC) -----------------------
__global__ void k_bn_prelu(float* __restrict__ h, const float* __restrict__ sums,
                           const float* __restrict__ sumsqs,
                           const float* __restrict__ gamma,
                           const float* __restrict__ beta,
                           const float* __restrict__ alpha, int M, int lC) {
  int idx = blockIdx.x * blockDim.x + threadIdx.x;
  if (idx >= (M << lC)) return;
  int c = idx & ((1 << lC) - 1);
  float invM = 1.0f / (float)M;
  float mean = sums[c] * invM;
  float var  = sumsqs[c] * invM - mean * mean;
  float v = (h[idx] - mean) * rsqrtf(var + 1e-5f) * gamma[c] + beta[c];
  float a = alpha[0];
  h[idx] = (v >= 0.f) ? v : a * v;
}

// ------------------ graph pooling scatter (C = 128) ------------------------
__global__ void k_pool_scatter(const float* __restrict__ h,
                               const int* __restrict__ batch,
                               float* __restrict__ pooled, float* __restrict__ cnt,
                               int M) {
  int idx = blockIdx.x * blockDim.x + threadIdx.x;
  if (idx >= (M << 7)) return;
  int m = idx >> 7, c = idx & 127;
  int g = batch[m];
  unsafeAtomicAdd(&pooled[(g << 7) + c], h[idx]);
  if (c == 0) unsafeAtomicAdd(&cnt[g], 1.0f);
}

__global__ void k_pool_div(float* __restrict__ pooled,
                           const float* __restrict__ cnt, int G) {
  int idx = blockIdx.x * blockDim.x + threadIdx.x;
  if (idx >= (G << 7)) return;
  pooled[idx] /= fmaxf(cnt[idx >> 7], 1.0f);
}

// ------------- split mu / exp(clip(logsigma))  (H = 1<<lH) -----------------
__global__ void k_split_exp(const float* __restrict__ z, float* __restrict__ mu,
                            float* __restrict__ sigma, int M, int lH) {
  int idx = blockIdx.x * blockDim.x + threadIdx.x;
  int H = 1 << lH;
  if (idx >= (M << (lH + 1))) return;
  int m = idx >> (lH + 1), c = idx & (2 * H - 1);
  float v = z[idx];
  if (c < H) {
    mu[(m << lH) + c] = v;
  } else {
    float t = fminf(fmaxf(v, -30.0f), 20.0f);
    sigma[(m << lH) + (c - H)] = expf(t);
  }
}

// ===========================================================================
extern "C" void kernel_launch(void* const* d_in, const int* in_sizes, int n_in,
                              void* d_out, int out_size, void* d_ws, size_t ws_size,
                              hipStream_t stream) {
  // ---- input decoding (setup_inputs dict order; params in JAX sorted-key
  //      tree-flatten order) ----
  const float* x     = (const float*)d_in[0];          // 20000x16
  const int*   eidx  = (const int*)d_in[1];            // 2x40000
  const int*   batch = (const int*)d_in[2];            // 20000
  const float* ea    = (const float*)d_in[3];          // 40000x4
  const int*   src   = eidx;
  const int*   dst   = eidx + NEDGES;
  auto F = [&](int i) { return (const float*)d_in[i]; };
  const float* bnbeta[4]  = {F(4), F(6), F(8), F(10)};
  const float* bngamma[4] = {F(5), F(7), F(9), F(11)};
  const float* Wedge[4] = {F(12), F(16), F(20), F(24)};
  const float* Wroot[4] = {F(13), F(17), F(21), F(25)};
  const float* bvec[4]  = {F(14), F(18), F(22), F(26)};
  const float* bedge[4] = {F(15), F(19), F(23), F(27)};
  const float* etaW = F(28); const float* etab = F(29);
  const float* poolW = F(30); const float* poolb = F(31);
  const float* prelu[4] = {F(32), F(33), F(34), F(35)};
  const float* zW1 = F(36); const float* zW2 = F(37);
  const float* zb1 = F(38); const float* zb2 = F(39);
  const float* zbnbeta = F(40); const float* zbngamma = F(41);
  const float* zprelu = F(42);

  // ---- workspace layout (floats) ----
  float* ws = (float*)d_ws;
  float* h0     = ws;                    // 20000*128
  float* h1     = h0 + NNODES * 128;     // 20000*128
  float* msg    = h1 + NNODES * 128;     // 40000*128
  float* agg    = msg + NEDGES * 128;    // 20000*128
  float* deg    = agg + NNODES * 128;    // 20000
  float* stats  = deg + NNODES;          // 256 (sum | sumsq)
  float* pooled = stats + 256;           // 1000*128
  float* cnt    = pooled + NGRAPH * 128; // 1000

  // ---- output layout: z_mu | z_sigma | eta_mu | eta_sigma ----
  float* out = (float*)d_out;
  float* out_zmu  = out;
  float* out_zsig = out + NNODES * 64;
  float* out_emu  = out + 2 * NNODES * 64;
  float* out_esig = out_emu + NGRAPH * 32;

  auto zero = [&](float* p, int n) {
    k_zero<<<(n + 255) / 256, 256, 0, stream>>>(p, n);
  };
  // shape-dispatched WMMA GEMM (all call-site shapes are known multiples)
  auto gemm = [&](const float* A, const float* B, const float* bias, float* D,
                  int M, int N, int K) {
    int gy_w = (N >= 64) ? 64 : N;       // slab width = min(N, 64)
    dim3 g((((M + 15) / 16) + 7) / 8, N / gy_w);
    if (K == 16 && N == 16)
      k_gemm<1, 16, true><<<g, 256, 0, stream>>>(A, B, bias, D, M, N);
    else if (K == 16 && N == 32)
      k_gemm<2, 16, true><<<g, 256, 0, stream>>>(A, B, bias, D, M, N);
    else if (K == 32)
      k_gemm<4, 32, true><<<g, 256, 0, stream>>>(A, B, bias, D, M, N);
    else if (K == 64)
      k_gemm<4, 64, true><<<g, 256, 0, stream>>>(A, B, bias, D, M, N);
    else if ((M & 15) == 0)
      k_gemm<4, 128, true><<<g, 256, 0, stream>>>(A, B, bias, D, M, N);
    else
      k_gemm<4, 128, false><<<g, 256, 0, stream>>>(A, B, bias, D, M, N);
  };

  const int couts[4] = {16, 32, 64, 128};
  const int cins[4]  = {16, 16, 32, 64};
  const int lCs[4]   = {4, 5, 6, 7};     // log2(cout)

  const float* hcur = x;   // layer input (starts at raw features, cin=16)
  float* hbuf[2] = {h0, h1};

  for (int l = 0; l < 4; ++l) {
    int cin = cins[l], cout = couts[l];
    float* hnext = hbuf[l & 1];

    // msg = A_aug @ W_aug   (E x cout)
    {
      int gy_w = (cout >= 64) ? 64 : cout;
      dim3 g(((NEDGES / 16) + 7) / 8, cout / gy_w);
      if (l == 0)
        k_ecc_msg<4, 1><<<g, 256, 0, stream>>>(hcur, src, ea, Wedge[l],
                                               bedge[l], msg, cout, NEDGES);
      else if (l == 1)
        k_ecc_msg<4, 2><<<g, 256, 0, stream>>>(hcur, src, ea, Wedge[l],
                                               bedge[l], msg, cout, NEDGES);
      else if (l == 2)
        k_ecc_msg<5, 4><<<g, 256, 0, stream>>>(hcur, src, ea, Wedge[l],
                                               bedge[l], msg, cout, NEDGES);
      else
        k_ecc_msg<6, 4><<<g, 256, 0, stream>>>(hcur, src, ea, Wedge[l],
                                               bedge[l], msg, cout, NEDGES);
    }
    // agg/deg scatter
    zero(agg, NNODES * cout);
    zero(deg, NNODES);
    {
      int n = NEDGES * cout;
      k_scatter_edges<<<(n + 255) / 256, 256, 0, stream>>>(msg, dst, agg, deg,
                                                           NEDGES, lCs[l]);
    }
    // hnext = hcur @ W_root + b ; += agg/deg
    gemm(hcur, Wroot[l], bvec[l], hnext, NNODES, cout, cin);
    {
      int n = NNODES * cout;
      k_add_agg<<<(n + 255) / 256, 256, 0, stream>>>(hnext, agg, deg, NNODES,
                                                     lCs[l]);
    }
    // BN + PReLU (in place)
    k_bn_stats<<<cout, 256, 0, stream>>>(hnext, NNODES, cout, stats, stats + 128);
    {
      int n = NNODES * cout;
      k_bn_prelu<<<(n + 255) / 256, 256, 0, stream>>>(
          hnext, stats, stats + 128, bngamma[l], bnbeta[l], prelu[l], NNODES,
          lCs[l]);
    }
    hcur = hnext;
  }
  const float* h = hcur;   // final node features, 20000x128 (in h1)

  // ---- z head ----
  gemm(h, zW1, zb1, msg, NNODES, 128, 128);                 // z1 in msg
  k_bn_stats<<<128, 256, 0, stream>>>(msg, NNODES, 128, stats, stats + 128);
  {
    int n = NNODES * 128;
    k_bn_prelu<<<(n + 255) / 256, 256, 0, stream>>>(
        msg, stats, stats + 128, zbngamma, zbnbeta, zprelu, NNODES, 7);
  }
  gemm(msg, zW2, zb2, agg, NNODES, 128, 128);               // z2 in agg
  {
    int n = NNODES * 128;
    k_split_exp<<<(n + 255) / 256, 256, 0, stream>>>(agg, out_zmu, out_zsig,
                                                     NNODES, 6);
  }

  // ---- eta head ----
  zero(pooled, NGRAPH * 128);
  zero(cnt, NGRAPH);
  {
    int n = NNODES * 128;
    k_pool_scatter<<<(n + 255) / 256, 256, 0, stream>>>(h, batch, pooled, cnt,
                                                        NNODES);
  }
  {
    int n = NGRAPH * 128;
    k_pool_div<<<(n + 255) / 256, 256, 0, stream>>>(pooled, cnt, NGRAPH);
  }
  gemm(pooled, poolW, poolb, msg, NGRAPH, 128, 128);        // p2 in msg
  gemm(msg, etaW, etab, agg, NGRAPH, 64, 128);              // eta in agg
  {
    int n = NGRAPH * 64;
    k_split_exp<<<(n + 255) / 256, 256, 0, stream>>>(agg, out_emu, out_esig,
                                                     NGRAPH, 5);
  }
}